// DiscreteDiffusion_23278722745101
// MI455X (gfx1250) — compile-verified
//
#include <hip/hip_runtime.h>
#include <hip/hip_bf16.h>

// Problem constants
#define B_DIM 64
#define S_DIM 4096
#define C_DIM 21
#define T_STEPS 200

typedef __attribute__((ext_vector_type(2))) float v2f;
typedef __attribute__((ext_vector_type(4))) float f4;
typedef __attribute__((ext_vector_type(8))) float v8f;

// ---------------------------------------------------------------------------
// Kernel 1: per-batch diffusion parameters.
// alpha_cumprod[t] = prod_{i=0..t} (1 - clip(1 - ac[i+1]/ac[i], 0, 0.999))
// ac[i] = cos^2(((i/200)+1e-4)/(1+1e-4) * pi/2)   (normalization by ac[0]
// cancels in the ratio). Stores {offdiag, diag, log_offdiag, log_diag}.
// ---------------------------------------------------------------------------
__device__ __forceinline__ float ac_fn(int i) {
    const float half_pi = 1.57079632679489662f;
    float s = ((float)i * (1.0f / (float)T_STEPS) + 1e-4f) / (1.0f + 1e-4f);
    float c = cosf(s * half_pi);
    return c * c;
}

__global__ void dd_params_kernel(const int* __restrict__ t, float* __restrict__ params) {
    int b = threadIdx.x;
    if (b >= B_DIM) return;
    int tb = t[b];
    float prod = 1.0f;
    float ac_prev = ac_fn(0);
    for (int i = 0; i <= tb; ++i) {
        float ac_next = ac_fn(i + 1);
        float beta = fminf(fmaxf(1.0f - ac_next / ac_prev, 0.0f), 0.999f);
        prod *= (1.0f - beta);
        ac_prev = ac_next;
    }
    float alpha = prod;
    float offv  = (1.0f - alpha) * (1.0f / (float)C_DIM);
    float diagv = alpha + offv;
    params[b * 4 + 0] = offv;
    params[b * 4 + 1] = diagv;
    params[b * 4 + 2] = logf(offv  + 1e-10f);
    params[b * 4 + 3] = logf(diagv + 1e-10f);
}

// ---------------------------------------------------------------------------
// Kernel 2: logits[b,s,d] = sum_c x0[b,s,c] * log_q[b,c,d] via
// V_WMMA_F32_16X16X4_F32 (exact f32, matching the f32 reference).
// One wave handles a 16-row S-tile. N=21 -> two 16-wide N tiles (cols >=21
// zero-padded / not stored). K=21 -> six K=4 chunks (k >= 21 zero-padded).
// B is synthesized in registers: log_q[k][d] = (k==d) ? log_diag : log_off.
//
// ISA VGPR layouts (CDNA5 ISA):
//   A 16x4 f32 : lane L<16 -> {A[L][k0], A[L][k0+1]}, lane L+16 -> {A[L][k0+2], A[L][k0+3]}
//   B 4x16 f32 : mirrored (row k striped across lanes within a VGPR)
//   C/D 16x16  : VGPR r -> row r (lanes 0-15, col=lane) / row r+8 (lanes 16-31)
// ---------------------------------------------------------------------------
__global__ __launch_bounds__(256) void dd_logits_wmma_kernel(
        const float* __restrict__ x0,
        const float* __restrict__ params,
        float* __restrict__ logits) {
    const int lane = threadIdx.x & 31;
    const int wave = threadIdx.x >> 5;
    const int b    = blockIdx.y;
    const int s0   = blockIdx.x * 128 + wave * 16;   // 8 waves * 16 rows = 128 rows/block

    const float log_off  = params[b * 4 + 2];
    const float log_diag = params[b * 4 + 3];

    const float* __restrict__ A = x0 + ((size_t)b * S_DIM + s0) * C_DIM;

    const int m     = lane & 15;          // row within tile (A) / column n (B, D)
    const int khalf = (lane >> 4) << 1;   // 0 for lanes 0-15, 2 for lanes 16-31

    v8f acc0 = {};   // columns 0..15
    v8f acc1 = {};   // columns 16..31 (only 16..20 stored)

#pragma unroll
    for (int kc = 0; kc < 6; ++kc) {
        const int k0 = kc * 4 + khalf;
        const int k1 = k0 + 1;

        v2f a;
        a.x = (k0 < C_DIM) ? A[m * C_DIM + k0] : 0.0f;
        a.y = (k1 < C_DIM) ? A[m * C_DIM + k1] : 0.0f;

        // B for N-tile 0: column d = m
        v2f b0;
        b0.x = (k0 < C_DIM) ? ((k0 == m) ? log_diag : log_off) : 0.0f;
        b0.y = (k1 < C_DIM) ? ((k1 == m) ? log_diag : log_off) : 0.0f;

        // B for N-tile 1: column d = m + 16 (zero beyond 21)
        const int n1 = m + 16;
        v2f b1;
        b1.x = (k0 < C_DIM && n1 < C_DIM) ? ((k0 == n1) ? log_diag : log_off) : 0.0f;
        b1.y = (k1 < C_DIM && n1 < C_DIM) ? ((k1 == n1) ? log_diag : log_off) : 0.0f;

        acc0 = __builtin_amdgcn_wmma_f32_16x16x4_f32(
                   false, a, false, b0, (short)0, acc0, false, false);
        acc1 = __builtin_amdgcn_wmma_f32_16x16x4_f32(
                   false, a, false, b1, (short)0, acc1, false, false);
    }

    // Store D: VGPR r holds row r (lanes 0-15) / row r+8 (lanes 16-31), col = lane%16.
    float* __restrict__ O = logits + ((size_t)b * S_DIM + s0) * C_DIM;
    const int rbase = (lane >> 4) * 8;
    const int col   = m;
    const bool hi_ok = (col + 16) < C_DIM;   // columns 16..20 only
#pragma unroll
    for (int r = 0; r < 8; ++r) {
        const int row = rbase + r;
        O[row * C_DIM + col] = acc0[r];
        if (hi_ok) O[row * C_DIM + col + 16] = acc1[r];
    }
}

// ---------------------------------------------------------------------------
// Kernel 3: q_t broadcast fill — 462 MB of 2-valued data. Pure streaming:
// one float4 non-temporal store per thread. Within a batch the (c,d) pattern
// repeats every 441 floats; diagonal iff (c*21+d) % 22 == 0. Per-batch block
// size S*441 = 1,806,336 is a multiple of 4, so a float4 never crosses a
// batch boundary. Exact grid, no guards.
// ---------------------------------------------------------------------------
__global__ __launch_bounds__(256) void dd_qt_fill_kernel(
        const float* __restrict__ params, float* __restrict__ qt) {
    const unsigned i = blockIdx.x * 256u + threadIdx.x;
    const unsigned e = i * 4u;                          // element index < 115,605,504 (fits u32)
    const unsigned b = e / (S_DIM * C_DIM * C_DIM);     // / 1,806,336
    const unsigned cd = e % (C_DIM * C_DIM);            // % 441 (valid: per-batch stride % 441 == 0)

    const float offv  = params[b * 4 + 0];
    const float diagv = params[b * 4 + 1];

    unsigned c0 = cd;
    unsigned c1 = cd + 1u; if (c1 >= 441u) c1 -= 441u;
    unsigned c2 = cd + 2u; if (c2 >= 441u) c2 -= 441u;
    unsigned c3 = cd + 3u; if (c3 >= 441u) c3 -= 441u;

    f4 v;
    v.x = (c0 % 22u == 0u) ? diagv : offv;
    v.y = (c1 % 22u == 0u) ? diagv : offv;
    v.z = (c2 % 22u == 0u) ? diagv : offv;
    v.w = (c3 % 22u == 0u) ? diagv : offv;

    __builtin_nontemporal_store(v, (f4*)(qt + e));
}

// ---------------------------------------------------------------------------
extern "C" void kernel_launch(void* const* d_in, const int* in_sizes, int n_in,
                              void* d_out, int out_size, void* d_ws, size_t ws_size,
                              hipStream_t stream) {
    const float* x0 = (const float*)d_in[0];   // [64, 4096, 21] f32
    const int*   t  = (const int*)d_in[1];     // [64] int32

    float* out    = (float*)d_out;
    float* logits = out;                                        // [64, 4096, 21]
    float* qt     = out + (size_t)B_DIM * S_DIM * C_DIM;        // [64, 4096, 21, 21]
    float* params = (float*)d_ws;                               // [64, 4] scratch

    // 1) per-batch alpha / log values
    dd_params_kernel<<<1, 64, 0, stream>>>(t, params);

    // 2) logits via WMMA f32 16x16x4: 8 waves/block * 16 rows = 128 rows/block
    dim3 lgrid(S_DIM / 128, B_DIM);   // (32, 64)
    dd_logits_wmma_kernel<<<lgrid, 256, 0, stream>>>(x0, params, logits);

    // 3) q_t streaming fill: 115,605,504 floats / 4 per thread / 256 per block
    const unsigned total4 = (unsigned)((size_t)B_DIM * S_DIM * C_DIM * C_DIM / 4); // 28,901,376
    dd_qt_fill_kernel<<<total4 / 256, 256, 0, stream>>>(params, qt);
}